// UsmProcess_sigma_43447889167044
// MI455X (gfx1250) — compile-verified
//
#include <hip/hip_runtime.h>

// ---------------------------------------------------------------------------
// Unsharp mask with per-batch Gaussian sigma, fused single-pass kernel.
// Both separable conv passes are banded-Toeplitz matmuls executed with
// V_WMMA_F32_16X16X4_F32 (fp32 WMMA, matches reference dtype).
// All WMMA work is wave-uniform straight-line code: EXEC is all-1s at every
// WMMA (ISA 7.12 requirement), no exec-masked loops around matrix ops.
// ---------------------------------------------------------------------------

#define B_      16
#define C_      3
#define H_      512
#define W_      896
#define RADIUS  12
#define TAPS    25           // 2*RADIUS+1
#define KBAND   40           // 16 (N) + 24 (band), multiple of WMMA K=4

// tile geometry (exact tiling: 896/64=14, 512/64=8)
#define TW      64           // output tile width  (4 col-subtiles)
#define TH      64           // output tile height (4 row-slabs of output)
#define IN_W    (TW + 2*RADIUS)          // 88
#define IN_H    96                       // 6 mid slabs * 16 covers TH + 2*RADIUS
#define IN_STRIDE  89        // odd -> stride-89 column fetch hits 16 distinct banks
#define MID_STRIDE 74        // 8*74%64=16, 2*74%64=20 -> D-store & B-fetch conflict-free

#define NSUB_H  24           // 6 slabs * 4 col-subtiles
#define NSUB_V  16           // 4 slabs * 4 col-subtiles
#define NWAVES  8

typedef float v2f __attribute__((ext_vector_type(2)));
typedef float v8f __attribute__((ext_vector_type(8)));

__device__ __forceinline__ int reflect_idx(int i, int n) {
    // jnp.pad mode='reflect' (no edge repeat); RADIUS < n so one fold suffices
    if (i < 0) i = -i;
    if (i >= n) i = 2 * n - 2 - i;
    return i;
}

__device__ __forceinline__ v8f wmma_f32_k4(v2f a, v2f b, v8f c) {
    return __builtin_amdgcn_wmma_f32_16x16x4_f32(
        /*neg_a=*/false, a, /*neg_b=*/false, b,
        /*c_mod=*/(short)0, c, /*reuse_a=*/false, /*reuse_b=*/false);
}

// --- kernel 1: per-batch normalized 25-tap Gaussian weights -> d_ws --------
__global__ void usm_weights_kernel(const float* __restrict__ param,
                                   float* __restrict__ wts) {
    int b = threadIdx.x;
    if (b >= B_) return;
    float inv_sigma = 1.0f / param[b];           // param shape (B,1)
    float w[TAPS];
    float s = 0.0f;
    #pragma unroll
    for (int i = 0; i < TAPS; ++i) {
        float x = (float)(i - RADIUS) * inv_sigma;
        w[i] = __expf(-0.5f * x * x);
        s += w[i];
    }
    float inv_s = 1.0f / s;
    #pragma unroll
    for (int i = 0; i < TAPS; ++i) wts[b * 32 + i] = w[i] * inv_s;
}

// --- kernel 2: fused reflect-pad blur (H then V via WMMA) + sharpen --------
__global__ __launch_bounds__(256)
void usm_fused_kernel(const float* __restrict__ img,
                      const float* __restrict__ param,
                      const float* __restrict__ wts,
                      float* __restrict__ out) {
    __shared__ float in_lds[IN_H * IN_STRIDE];    // 34.2 KB halo'd input tile
    __shared__ float mid_lds[IN_H * MID_STRIDE];  // 28.4 KB h-blurred intermediate
    __shared__ float band[KBAND * 16];            // banded Toeplitz weight matrix
    __shared__ float wlds[TAPS];

    const int tid  = threadIdx.x;
    const int lane = tid & 31;
    const int wv   = tid >> 5;          // wave id 0..7
    const int m    = lane & 15;         // row (A-role) / col (B-role) index
    const int hi   = lane >> 4;         // selects K-pair {2hi, 2hi+1}

    const int bc = blockIdx.z;          // 0..B*C-1
    const int b  = bc / C_;
    const int x0 = blockIdx.x * TW;
    const int y0 = blockIdx.y * TH;
    const size_t base = (size_t)bc * (size_t)(H_ * W_);

    // per-batch taps
    if (tid < TAPS) wlds[tid] = wts[b * 32 + tid];
    __syncthreads();

    // band[j][n] = w[j-n] if 0<=j-n<TAPS else 0   (j:0..39, n:0..15)
    for (int idx = tid; idx < KBAND * 16; idx += 256) {
        int j = idx >> 4, n = idx & 15;
        int d = j - n;
        band[idx] = (d >= 0 && d < TAPS) ? wlds[d] : 0.0f;
    }

    // cooperative halo'd tile load with reflect padding
    for (int idx = tid; idx < IN_H * IN_W; idx += 256) {
        int r  = idx / IN_W, c = idx % IN_W;
        int gr = reflect_idx(y0 - RADIUS + r, H_);
        int gc = reflect_idx(x0 - RADIUS + c, W_);
        in_lds[r * IN_STRIDE + c] = img[base + (size_t)gr * W_ + gc];
    }
    __syncthreads();

    // ---- horizontal pass: Mid(16x16) = In(16x40) x Band(40x16) ----
    // 24 subtiles over 8 waves -> exactly 3 per wave, fully uniform/unrolled.
    #pragma unroll
    for (int i = 0; i < NSUB_H / NWAVES; ++i) {
        const int st   = wv + NWAVES * i;   // 0..23, wave-uniform
        const int slab = st >> 2;           // 0..5
        const int col0 = (st & 3) * 16;     // 0,16,32,48
        const int row0 = slab * 16;
        v8f acc = {0.f,0.f,0.f,0.f,0.f,0.f,0.f,0.f};
        #pragma unroll
        for (int k0 = 0; k0 < KBAND; k0 += 4) {
            v2f a, bm;
            const float* arow = &in_lds[(row0 + m) * IN_STRIDE + col0 + k0 + 2 * hi];
            a.x  = arow[0];
            a.y  = arow[1];
            bm.x = band[(k0 + 2 * hi)     * 16 + m];
            bm.y = band[(k0 + 2 * hi + 1) * 16 + m];
            acc = wmma_f32_k4(a, bm, acc);
        }
        #pragma unroll
        for (int v = 0; v < 8; ++v) {
            int mm = v + 8 * hi;        // D-layout: VGPR v holds M=v (lo) / M=8+v (hi)
            mid_lds[(row0 + mm) * MID_STRIDE + col0 + m] = acc[v];
        }
    }
    __syncthreads();

    // ---- vertical pass + sharpen: Out(16x16) = BandT(16x40) x Mid(40x16) ----
    // 16 subtiles over 8 waves -> exactly 2 per wave, fully uniform/unrolled.
    const float p = param[b];
    #pragma unroll
    for (int i = 0; i < NSUB_V / NWAVES; ++i) {
        const int st   = wv + NWAVES * i;   // 0..15, wave-uniform
        const int slab = st >> 2;           // 0..3
        const int col0 = (st & 3) * 16;     // 0,16,32,48
        v8f acc = {0.f,0.f,0.f,0.f,0.f,0.f,0.f,0.f};
        #pragma unroll
        for (int k0 = 0; k0 < KBAND; k0 += 4) {
            v2f a, bm;
            a.x  = band[(k0 + 2 * hi)     * 16 + m];   // Av[m][j] = band[j][m]
            a.y  = band[(k0 + 2 * hi + 1) * 16 + m];
            bm.x = mid_lds[(slab * 16 + k0 + 2 * hi)     * MID_STRIDE + col0 + m];
            bm.y = mid_lds[(slab * 16 + k0 + 2 * hi + 1) * MID_STRIDE + col0 + m];
            acc = wmma_f32_k4(a, bm, acc);
        }
        #pragma unroll
        for (int v = 0; v < 8; ++v) {
            int mm  = v + 8 * hi;
            int gr  = y0 + slab * 16 + mm;
            int gc  = x0 + col0 + m;
            float orig = in_lds[(RADIUS + slab * 16 + mm) * IN_STRIDE +
                                (RADIUS + col0 + m)];
            out[base + (size_t)gr * W_ + gc] = (orig - acc[v]) * p + orig;
        }
    }
}

extern "C" void kernel_launch(void* const* d_in, const int* in_sizes, int n_in,
                              void* d_out, int out_size, void* d_ws, size_t ws_size,
                              hipStream_t stream) {
    const float* img   = (const float*)d_in[0];   // (16,3,512,896) f32
    const float* param = (const float*)d_in[1];   // (16,1) f32
    float*       out   = (float*)d_out;
    float*       wts   = (float*)d_ws;            // B x 32 floats of scratch

    usm_weights_kernel<<<1, 32, 0, stream>>>(param, wts);

    dim3 grid(W_ / TW, H_ / TH, B_ * C_);         // 14 x 8 x 48
    usm_fused_kernel<<<grid, 256, 0, stream>>>(img, param, wts, out);
}